// SModule_12592844112143
// MI455X (gfx1250) — compile-verified
//
#include <hip/hip_runtime.h>
#include <hip/hip_bf16.h>
#include <math.h>

#define BB 2
#define SS 2048
#define DD 256
#define WW 64
#define RR 64
#define LL 2
#define HH 512
#define NSEQ (SS + 1)
#define MT (BB * NSEQ)
#define LNEPS 1e-5f
#define ATT_SCALE 0.125f   // 1/sqrt(R=64)

typedef float v2f __attribute__((ext_vector_type(2)));
typedef float v8f __attribute__((ext_vector_type(8)));
typedef int   v4i __attribute__((vector_size(16)));   // int4, matches builtin proto

// CDNA5 async global->LDS staging (ASYNCcnt-tracked), with sync fallback.
#if defined(__has_builtin)
# if __has_builtin(__builtin_amdgcn_global_load_async_to_lds_b128)
#  define HAVE_ASYNC_LDS 1
# endif
#endif

#ifdef HAVE_ASYNC_LDS
# define GLOBAL_V4I(p) ((__attribute__((address_space(1))) v4i*)(p))
# define LDS_V4I(p)    ((__attribute__((address_space(3))) v4i*)(p))
# if __has_builtin(__builtin_amdgcn_s_wait_asynccnt)
#  define ASYNC_WAIT0() __builtin_amdgcn_s_wait_asynccnt(0)
# else
#  define ASYNC_WAIT0() asm volatile("s_wait_asynccnt 0x0" ::: "memory")
# endif
#else
# define ASYNC_WAIT0() ((void)0)
#endif

__device__ __forceinline__ float gelu_t(float x) {
    // jax.nn.gelu default (approximate=True, tanh form)
    float u = 0.7978845608028654f * (x + 0.044715f * x * x * x);
    return 0.5f * x * (1.f + tanhf(u));
}

// ---------------------------------------------------------------------------
// Embedding + input layernorm + state projection. One block per token row.
// Extra BB blocks at the end write the anchor rows (n = 0).
// ---------------------------------------------------------------------------
__global__ __launch_bounds__(256) void k_embed(
    const int* __restrict__ ids, const float* __restrict__ emb,
    const float* __restrict__ pos, const float* __restrict__ lnw,
    const float* __restrict__ lnb, const float* __restrict__ spw,
    const float* __restrict__ spb, const float* __restrict__ av,
    const float* __restrict__ asc, float* __restrict__ val,
    float* __restrict__ state)
{
    __shared__ float red[DD];
    const int blk = blockIdx.x;
    const int t = threadIdx.x;
    if (blk >= BB * SS) {
        const int b = blk - BB * SS;
        val[((size_t)b * NSEQ) * DD + t] = av[t];
        if (t == 0) state[(size_t)b * NSEQ] = asc[0];
        return;
    }
    const int b = blk / SS, s = blk % SS;
    const int id = ids[b * SS + s];
    float x = emb[(size_t)id * DD + t] + pos[(size_t)s * DD + t];
    red[t] = x; __syncthreads();
    for (int st = 128; st > 0; st >>= 1) { if (t < st) red[t] += red[t + st]; __syncthreads(); }
    const float m = red[0] * (1.f / DD); __syncthreads();
    const float xm = x - m;
    red[t] = xm * xm; __syncthreads();
    for (int st = 128; st > 0; st >>= 1) { if (t < st) red[t] += red[t + st]; __syncthreads(); }
    const float var = red[0] * (1.f / DD); __syncthreads();
    const float y = xm * rsqrtf(var + LNEPS) * lnw[t] + lnb[t];
    val[((size_t)b * NSEQ + (s + 1)) * DD + t] = y;
    red[t] = y * spw[t]; __syncthreads();
    for (int st = 128; st > 0; st >>= 1) { if (t < st) red[t] += red[t + st]; __syncthreads(); }
    if (t == 0) state[(size_t)b * NSEQ + (s + 1)] = red[0] + spb[0];
}

// ---------------------------------------------------------------------------
// Row layernorm: Y[row] = LN(X[row]) * w + b. One block (256 thr) per row.
// ---------------------------------------------------------------------------
__global__ __launch_bounds__(256) void k_ln(
    const float* __restrict__ X, const float* __restrict__ w,
    const float* __restrict__ bprm, float* __restrict__ Y)
{
    __shared__ float red[DD];
    const int row = blockIdx.x;
    const int t = threadIdx.x;
    const float v = X[(size_t)row * DD + t];
    red[t] = v; __syncthreads();
    for (int st = 128; st > 0; st >>= 1) { if (t < st) red[t] += red[t + st]; __syncthreads(); }
    const float m = red[0] * (1.f / DD); __syncthreads();
    const float xm = v - m;
    red[t] = xm * xm; __syncthreads();
    for (int st = 128; st > 0; st >>= 1) { if (t < st) red[t] += red[t + st]; __syncthreads(); }
    const float var = red[0] * (1.f / DD);
    Y[(size_t)row * DD + t] = xm * rsqrtf(var + LNEPS) * w[t] + bprm[t];
}

// ---------------------------------------------------------------------------
// WMMA f32 GEMM: C[M,Nn] = epi(A[M,K] @ Bm[K,Nn] + bias) (+ Res)
// 64x64 C tile / 256-thread block, 8 waves, each wave owns a 16x32 sub-tile
// as two V_WMMA_F32_16X16X4_F32 chains. Double-buffered BK=64 LDS staging via
// GLOBAL_LOAD_ASYNC_TO_LDS_B128 (ASYNCcnt) when available.
// A-tile rows past M are clamped to row M-1: they only feed C rows that are
// never stored, so no zero-fill / EXEC masking is needed (WMMA wants EXEC=~0).
// act: 0 = none, 1 = tanh-gelu. Res may alias C (element-wise read->write).
// ---------------------------------------------------------------------------
#define BM 64
#define BN 64
#define BK 64
#define LDA 68
#define LDB 68

__global__ __launch_bounds__(256) void k_gemm(
    const float* __restrict__ A, const float* __restrict__ Bm,
    const float* __restrict__ bias, const float* __restrict__ Res,
    float* __restrict__ C, int M, int K, int Nn, int act)
{
    __shared__ float As[2][BM * LDA];
    __shared__ float Bs[2][BK * LDB];
    const int tid  = threadIdx.x;
    const int lane = tid & 31;
    const int wave = tid >> 5;
    const int lh   = lane >> 4;   // half-wave: selects K pair / M+8 per ISA layout
    const int ll   = lane & 15;
    const int m0 = blockIdx.y * BM;
    const int n0 = blockIdx.x * BN;
    const int wm = (wave & 3) * 16;
    const int wn = (wave >> 2) * 32;

    v8f acc0 = {0.f, 0.f, 0.f, 0.f, 0.f, 0.f, 0.f, 0.f};
    v8f acc1 = acc0;

    // Stage one BK-chunk of A and B tiles into LDS buffer `buf`.
    auto stage = [&](int buf, int k0) {
#pragma unroll
        for (int i = 0; i < 4; ++i) {                 // A tile: 64x64 floats
            const int slot = tid + i * 256;           // 1024 float4 slots
            const int r = slot >> 4;
            const int c = (slot & 15) << 2;
            int gm = m0 + r; if (gm >= M) gm = M - 1; // clamp (rows never stored)
            const float* src = A + (size_t)gm * K + (k0 + c);
            float* dst = &As[buf][r * LDA + c];
#ifdef HAVE_ASYNC_LDS
            __builtin_amdgcn_global_load_async_to_lds_b128(
                GLOBAL_V4I(src), LDS_V4I(dst), 0, 0);
#else
            float4 v = *(const float4*)src;
            dst[0] = v.x; dst[1] = v.y; dst[2] = v.z; dst[3] = v.w;
#endif
            if (k0 + 2 * BK < K) __builtin_prefetch(src + 2 * BK, 0, 1);
        }
#pragma unroll
        for (int i = 0; i < 4; ++i) {                 // B tile: 64x64 floats
            const int slot = tid + i * 256;
            const int r = slot >> 4;
            const int c = (slot & 15) << 2;
            const float* src = Bm + (size_t)(k0 + r) * Nn + (n0 + c);
            float* dst = &Bs[buf][r * LDB + c];
#ifdef HAVE_ASYNC_LDS
            __builtin_amdgcn_global_load_async_to_lds_b128(
                GLOBAL_V4I(src), LDS_V4I(dst), 0, 0);
#else
            float4 v = *(const float4*)src;
            dst[0] = v.x; dst[1] = v.y; dst[2] = v.z; dst[3] = v.w;
#endif
            if (k0 + 2 * BK < K) __builtin_prefetch(src + (size_t)2 * BK * Nn, 0, 1);
        }
    };

    auto compute = [&](int buf) {
#pragma unroll
        for (int kk = 0; kk < BK; kk += 4) {
            // A frag (16x4): lanes 0-15 -> K=kk+0,kk+1 ; lanes 16-31 -> K=kk+2,kk+3
            v2f a;
            const float* ap = &As[buf][(wm + ll) * LDA + kk + 2 * lh];
            a.x = ap[0]; a.y = ap[1];
            // B frag (4x16): vgpr0 row K=kk+2*lh, vgpr1 row K=kk+1+2*lh, col = ll
            v2f b0, b1;
            const float* bp0 = &Bs[buf][(kk + 2 * lh) * LDB + wn + ll];
            b0.x = bp0[0];   b0.y = bp0[LDB];
            const float* bp1 = bp0 + 16;
            b1.x = bp1[0];   b1.y = bp1[LDB];
            acc0 = __builtin_amdgcn_wmma_f32_16x16x4_f32(false, a, false, b0,
                                                         (short)0, acc0, false, false);
            acc1 = __builtin_amdgcn_wmma_f32_16x16x4_f32(false, a, false, b1,
                                                         (short)0, acc1, false, false);
        }
    };

    const int nk = K / BK;
    int buf = 0;
    stage(0, 0);
    ASYNC_WAIT0();
    __syncthreads();
    for (int c = 0; c < nk; ++c) {
        if (c + 1 < nk) stage(buf ^ 1, (c + 1) * BK);   // overlap with compute
        compute(buf);
        ASYNC_WAIT0();        // next buffer landed (this wave's async loads)
        __syncthreads();      // all waves done reading buf + done staging buf^1
        buf ^= 1;
    }

    // C layout: vgpr v -> row m0+wm+v+8*lh, col n0+wn+ll (and +16 for acc1)
#pragma unroll
    for (int v = 0; v < 8; ++v) {
        const int gm = m0 + wm + v + 8 * lh;
        if (gm < M) {
            const int gn0 = n0 + wn + ll;
            float x0 = acc0[v];
            float x1 = acc1[v];
            if (bias) { x0 += bias[gn0]; x1 += bias[gn0 + 16]; }
            if (act == 1) { x0 = gelu_t(x0); x1 = gelu_t(x1); }
            if (Res) { x0 += Res[(size_t)gm * Nn + gn0]; x1 += Res[(size_t)gm * Nn + gn0 + 16]; }
            C[(size_t)gm * Nn + gn0]      = x0;
            C[(size_t)gm * Nn + gn0 + 16] = x1;
        }
    }
}

// ---------------------------------------------------------------------------
// Windowed signed-softmax attention. One block (8 wave32s) per (b, n).
// e_w = q[n].k[n-w]/sqrt(R); w = sign(e)*softmax(|e| masked); delta sums.
// Writes val_tmp = val + delta_val (pre-LN), s_tmp = state + delta_state.
// ---------------------------------------------------------------------------
__global__ __launch_bounds__(256) void k_attn(
    const float* __restrict__ q, const float* __restrict__ k,
    const float* __restrict__ val, const float* __restrict__ state,
    float* __restrict__ val_tmp, float* __restrict__ s_tmp)
{
    __shared__ float e_sh[WW + 1];
    __shared__ float wt_sh[WW + 1];
    __shared__ float red[256];
    const int bn = blockIdx.x;
    const int b = bn / NSEQ;
    const int n = bn % NSEQ;
    const int t = threadIdx.x;
    const int wave = t >> 5;
    const int lane = t & 31;
    const int wmax = n < WW ? n : WW;
    const float* qp = q + ((size_t)b * NSEQ + n) * RR;

    for (int w = wave; w <= WW; w += 8) {
        float p = 0.f;
        if (w <= wmax) {
            const float* kp = k + ((size_t)b * NSEQ + (n - w)) * RR;
            p = qp[lane] * kp[lane] + qp[lane + 32] * kp[lane + 32];
        }
        for (int off = 16; off > 0; off >>= 1) p += __shfl_xor(p, off, 32);
        if (lane == 0) e_sh[w] = p * ATT_SCALE;
    }
    __syncthreads();

    float e = 0.f; bool valid = false;
    if (t <= WW) { e = e_sh[t]; valid = (t <= wmax); }
    const float a = valid ? fabsf(e) : -3.4e38f;
    red[t] = a; __syncthreads();
    for (int st = 128; st > 0; st >>= 1) { if (t < st) red[t] = fmaxf(red[t], red[t + st]); __syncthreads(); }
    const float amax = red[0]; __syncthreads();
    const float ex = valid ? expf(a - amax) : 0.f;
    red[t] = ex; __syncthreads();
    for (int st = 128; st > 0; st >>= 1) { if (t < st) red[t] += red[t + st]; __syncthreads(); }
    const float denom = red[0];
    if (t <= WW) {
        const float sg = (e > 0.f ? 1.f : 0.f) - (e < 0.f ? 1.f : 0.f);
        wt_sh[t] = valid ? sg * ex / denom : 0.f;
    }
    __syncthreads();

    const size_t rowbase = ((size_t)b * NSEQ + n) * DD;
    const float* vb = val + ((size_t)b * NSEQ) * DD;
    float dv = 0.f;
    for (int w = 0; w <= wmax; ++w) dv += wt_sh[w] * vb[(size_t)(n - w) * DD + t];
    val_tmp[rowbase + t] = val[rowbase + t] + dv;

    if (t == 0) {
        const float* sb = state + (size_t)b * NSEQ;
        float ds = 0.f;
        for (int w = 0; w <= wmax; ++w) ds += wt_sh[w] * sb[n - w];
        s_tmp[(size_t)b * NSEQ + n] = sb[n] + ds;
    }
}

// ---------------------------------------------------------------------------
// Global signed softmax over the N axis of state: one block per batch.
// ---------------------------------------------------------------------------
__global__ __launch_bounds__(512) void k_state_softmax(
    const float* __restrict__ s_in, float* __restrict__ s_out)
{
    __shared__ float red[512];
    const int b = blockIdx.x;
    const int t = threadIdx.x;
    const float* x = s_in + (size_t)b * NSEQ;
    float m = -3.4e38f;
    for (int i = t; i < NSEQ; i += 512) m = fmaxf(m, fabsf(x[i]));
    red[t] = m; __syncthreads();
    for (int st = 256; st > 0; st >>= 1) { if (t < st) red[t] = fmaxf(red[t], red[t + st]); __syncthreads(); }
    const float mx = red[0]; __syncthreads();
    float s = 0.f;
    for (int i = t; i < NSEQ; i += 512) s += expf(fabsf(x[i]) - mx);
    red[t] = s; __syncthreads();
    for (int st = 256; st > 0; st >>= 1) { if (t < st) red[t] += red[t + st]; __syncthreads(); }
    const float denom = red[0];
    for (int i = t; i < NSEQ; i += 512) {
        const float v = x[i];
        const float sg = (v > 0.f ? 1.f : 0.f) - (v < 0.f ? 1.f : 0.f);
        s_out[(size_t)b * NSEQ + i] = sg * expf(fabsf(v) - mx) / denom;
    }
}

// ---------------------------------------------------------------------------
// Pack (state, val) into d_out.
// ---------------------------------------------------------------------------
__global__ __launch_bounds__(256) void k_out(
    const float* __restrict__ state, const float* __restrict__ val,
    float* __restrict__ out)
{
    const size_t ns = (size_t)MT;
    const size_t tot = ns + ns * (size_t)DD;
    for (size_t i = blockIdx.x * (size_t)blockDim.x + threadIdx.x; i < tot;
         i += (size_t)gridDim.x * blockDim.x)
        out[i] = (i < ns) ? state[i] : val[i - ns];
}

// ---------------------------------------------------------------------------
extern "C" void kernel_launch(void* const* d_in, const int* in_sizes, int n_in,
                              void* d_out, int out_size, void* d_ws, size_t ws_size,
                              hipStream_t stream)
{
    const int*   ids = (const int*)d_in[0];
    const float* emb = (const float*)d_in[1];
    const float* pos = (const float*)d_in[2];
    const float* ilw = (const float*)d_in[3];
    const float* ilb = (const float*)d_in[4];
    const float* spw = (const float*)d_in[5];
    const float* spb = (const float*)d_in[6];
    const float* av  = (const float*)d_in[7];
    const float* asc = (const float*)d_in[8];
    const float* U   = (const float*)d_in[9];
    const float* V   = (const float*)d_in[10];
    const float* lnw = (const float*)d_in[11];
    const float* lnb = (const float*)d_in[12];
    const float* W1  = (const float*)d_in[13];
    const float* b1  = (const float*)d_in[14];
    const float* W2  = (const float*)d_in[15];
    const float* b2  = (const float*)d_in[16];
    float* out = (float*)d_out;

    float* ws = (float*)d_ws;
    size_t off = 0;
    float* val     = ws + off; off += (size_t)MT * DD;   // current activations
    float* val_tmp = ws + off; off += (size_t)MT * DD;   // val + delta (pre-LN)
    float* h       = ws + off; off += (size_t)MT * HH;   // MLP hidden
    float* q       = ws + off; off += (size_t)MT * RR;
    float* k       = ws + off; off += (size_t)MT * RR;
    float* state   = ws + off; off += (size_t)MT;
    float* s_tmp   = ws + off; off += (size_t)MT;
    (void)in_sizes; (void)n_in; (void)out_size; (void)ws_size;

    k_embed<<<BB * SS + BB, 256, 0, stream>>>(ids, emb, pos, ilw, ilb, spw, spb,
                                              av, asc, val, state);

    const int gy = (MT + BM - 1) / BM;   // 65
    for (int l = 0; l < LL; ++l) {
        const float* Ul = U + (size_t)l * DD * RR;
        const float* Vl = V + (size_t)l * DD * RR;
        k_gemm<<<dim3(RR / BN, gy), 256, 0, stream>>>(val, Ul, nullptr, nullptr,
                                                      q, MT, DD, RR, 0);
        k_gemm<<<dim3(RR / BN, gy), 256, 0, stream>>>(val, Vl, nullptr, nullptr,
                                                      k, MT, DD, RR, 0);
        k_attn<<<BB * NSEQ, 256, 0, stream>>>(q, k, val, state, val_tmp, s_tmp);
        k_state_softmax<<<BB, 512, 0, stream>>>(s_tmp, state);
        k_ln<<<MT, 256, 0, stream>>>(val_tmp, lnw + (size_t)l * DD,
                                     lnb + (size_t)l * DD, val);
        k_gemm<<<dim3(HH / BN, gy), 256, 0, stream>>>(val, W1 + (size_t)l * DD * HH,
                                                      b1 + (size_t)l * HH, nullptr,
                                                      h, MT, DD, HH, 1);
        k_gemm<<<dim3(DD / BN, gy), 256, 0, stream>>>(h, W2 + (size_t)l * HH * DD,
                                                      b2 + (size_t)l * DD, val,
                                                      val, MT, HH, DD, 0);
    }
    const size_t tot = (size_t)MT + (size_t)MT * DD;
    k_out<<<(int)((tot + 255) / 256), 256, 0, stream>>>(state, val, out);
}